// TagOffsetReguLoss_87136296501797
// MI455X (gfx1250) — compile-verified
//
#include <hip/hip_runtime.h>

// Problem constants (from reference)
#define BB   16
#define TAGS 8
#define HH   320
#define WW2  320
#define NN   8192
#define GG   128
#define KK   64          // N / G
#define REGU_THR 0.001f

typedef __attribute__((ext_vector_type(2))) float v2f;
typedef __attribute__((ext_vector_type(8))) float v8f;

// pull contribution scale: 1 / (K * (G+EPS) * B)
__device__ __constant__ float PULL_SCALE = (float)(1.0 / (64.0 * (128.0 + 1e-6) * 16.0));

// ---------------- zero accumulators ----------------
__global__ void zero_acc_kernel(float* acc) {
    if (threadIdx.x < 2) acc[threadIdx.x] = 0.0f;
}

// ---------------- fused: bilinear sample + group tag + pull ----------------
// One 64-thread block (2 waves) per (b,g) group; thread k handles sample k.
// feats never leave registers; tags reduced via LDS (deterministic order),
// pull contribution block-reduced and accumulated once per block.
__global__ void __launch_bounds__(KK)
fused_sample_pull_kernel(const float* __restrict__ pred,
                         const float* __restrict__ anchor,
                         const float* __restrict__ offp,
                         float* __restrict__ tags,
                         float* __restrict__ acc) {
    int bg = blockIdx.x;            // b*GG + g
    int b  = bg / GG;
    int g  = bg % GG;
    int k  = threadIdx.x;           // 0..63
    int n  = g * KK + k;
    int idx = b * NN + n;

    // ---- bilinear sample (8 channels, zero padding, align_corners=False) ----
    float ax = 0.5f * (anchor[4*n + 0] + anchor[4*n + 2]);
    float ay = 0.5f * (anchor[4*n + 1] + anchor[4*n + 3]);
    float cx = ax + offp[2*idx + 0];
    float cy = ay + offp[2*idx + 1];

    float x0f = floorf(cx), y0f = floorf(cy);
    float wx1 = cx - x0f, wx0 = 1.0f - wx1;
    float wy1 = cy - y0f, wy0 = 1.0f - wy1;

    const float* img = pred + (size_t)b * TAGS * HH * WW2;
    float f[TAGS];
#pragma unroll
    for (int t = 0; t < TAGS; ++t) f[t] = 0.0f;

#pragma unroll
    for (int c = 0; c < 4; ++c) {
        float xf = x0f + (float)(c & 1);
        float yf = y0f + (float)(c >> 1);
        float w  = ((c & 1) ? wx1 : wx0) * ((c >> 1) ? wy1 : wy0);
        bool inb = (xf >= 0.0f) && (xf <= (float)(WW2 - 1)) &&
                   (yf >= 0.0f) && (yf <= (float)(HH  - 1));
        int xi = (int)fminf(fmaxf(xf, 0.0f), (float)(WW2 - 1));
        int yi = (int)fminf(fmaxf(yf, 0.0f), (float)(HH  - 1));
        float wm = inb ? w : 0.0f;
        size_t base = (size_t)yi * WW2 + xi;
#pragma unroll
        for (int t = 0; t < TAGS; ++t)
            f[t] += wm * img[(size_t)t * HH * WW2 + base];
    }

    // ---- group tag = mean over K, via LDS (deterministic order) ----
    __shared__ float sFeat[TAGS * KK];   // [t][k]
    __shared__ float sTag[TAGS];
    __shared__ float sWave[2];
#pragma unroll
    for (int t = 0; t < TAGS; ++t) sFeat[t * KK + k] = f[t];
    __syncthreads();
    if (k < TAGS) {
        float s = 0.0f;
        for (int kk = 0; kk < KK; ++kk) s += sFeat[k * KK + kk];
        float tg = s * (1.0f / KK);
        sTag[k] = tg;
        tags[(size_t)bg * TAGS + k] = tg;     // for the push kernel
    }
    __syncthreads();

    // ---- pull contribution ----
    float s = 0.0f;
#pragma unroll
    for (int t = 0; t < TAGS; ++t) { float d = f[t] - sTag[t]; s += d * d; }
    s *= (1.0f / TAGS);
    if (s < REGU_THR) s = 0.0f;

#pragma unroll
    for (int off = 16; off > 0; off >>= 1)
        s += __shfl_down(s, off, 32);
    if ((k & 31) == 0) sWave[k >> 5] = s;
    __syncthreads();
    if (k == 0) atomicAdd(&acc[1], (sWave[0] + sWave[1]) * PULL_SCALE);
}

// ---------------- push loss: WMMA Gram matrix + IoU mask + reduce ----------------
// One wave32 per 16x16 tile of dot = tags * tags^T (inner dim 8).
// sq / expanded boxes / areas for the tile's 32 rows staged through LDS.
__global__ void __launch_bounds__(32)
push_wmma_kernel(const float* __restrict__ tags,
                 const float* __restrict__ gtb,
                 float* __restrict__ acc) {
    int b    = blockIdx.x >> 6;
    int tile = blockIdx.x & 63;
    int i0 = (tile >> 3) * 16;
    int j0 = (tile &  7) * 16;
    int lane = threadIdx.x;          // 0..31, full wave, no divergence anywhere
    int half = lane >> 4;            // 0 or 1
    int l    = lane & 15;

    const float* tgb = tags + (size_t)b * GG * TAGS;

    // ---- per-row staging: slots [0..15] = rows i0+.., [16..31] = rows j0+.. ----
    __shared__ float sSq[32];
    __shared__ float sBox[32 * 4];
    __shared__ float sArea[32];
    {
        int row = (half == 0) ? (i0 + l) : (j0 + l);
        float sqv = 0.0f;
#pragma unroll
        for (int t = 0; t < TAGS; ++t) {
            float v = tgb[row * TAGS + t];
            sqv += v * v;
        }
        sSq[lane] = sqv;
        const float* bx = gtb + (size_t)(b * GG + row) * 4;
        float x1 = bx[0], y1 = bx[1], x2 = bx[2], y2 = bx[3];
        float w = x2 - x1 + 1.0f, h = y2 - y1 + 1.0f;
        float ex1 = x1 - 0.25f * w, ey1 = y1 - 0.25f * h;
        float ex2 = x2 + 0.25f * w, ey2 = y2 + 0.25f * h;
        sBox[lane * 4 + 0] = ex1; sBox[lane * 4 + 1] = ey1;
        sBox[lane * 4 + 2] = ex2; sBox[lane * 4 + 3] = ey2;
        sArea[lane] = (ex2 - ex1 + 1.0f) * (ey2 - ey1 + 1.0f);
    }
    __syncthreads();

    // ---- Gram tile via f32 WMMA, K = 8 as two K=4 steps ----
    // A 16x4 layout: lanes0-15 {v0:K0,v1:K1}, lanes16-31 {v0:K2,v1:K3}.
    // B 4x16 mirrors it; for the Gram case B-frag(j0) == A-frag(j0).
    v8f d = {0.f, 0.f, 0.f, 0.f, 0.f, 0.f, 0.f, 0.f};
#pragma unroll
    for (int kk = 0; kk < 2; ++kk) {
        int kbase = 4 * kk + 2 * half;
        v2f a, bm;
        a.x  = tgb[(i0 + l) * TAGS + kbase + 0];
        a.y  = tgb[(i0 + l) * TAGS + kbase + 1];
        bm.x = tgb[(j0 + l) * TAGS + kbase + 0];
        bm.y = tgb[(j0 + l) * TAGS + kbase + 1];
        d = __builtin_amdgcn_wmma_f32_16x16x4_f32(false, a, false, bm,
                                                  (short)0, d, false, false);
    }

    // ---- D layout: lanes0-15 -> M=r, N=lane; lanes16-31 -> M=8+r, N=lane-16 ----
    int sj = 16 + l;                 // LDS slot of column j
    float sqj = sSq[sj];
    float jx1 = sBox[sj*4 + 0], jy1 = sBox[sj*4 + 1];
    float jx2 = sBox[sj*4 + 2], jy2 = sBox[sj*4 + 3];
    float aj  = sArea[sj];

    float lsum = 0.0f;
#pragma unroll
    for (int r = 0; r < 8; ++r) {
        int si = 8 * half + r;       // LDS slot of row i
        float diff = (sSq[si] + sqj - 2.0f * d[r]) * (1.0f / TAGS);
        float p    = expf(-diff);
        float ix1 = sBox[si*4 + 0], iy1 = sBox[si*4 + 1];
        float ix2 = sBox[si*4 + 2], iy2 = sBox[si*4 + 3];
        float iw = fmaxf(fminf(ix2, jx2) - fmaxf(ix1, jx1) + 1.0f, 0.0f);
        float ih = fmaxf(fminf(iy2, jy2) - fmaxf(iy1, jy1) + 1.0f, 0.0f);
        float inter = iw * ih;
        float iou   = inter / (sArea[si] + aj - inter);
        float pw = (iou > 0.0f) ? 1.0f : 0.1f;
        if (p < REGU_THR) pw = 0.0f;
        lsum += p * pw;
    }
#pragma unroll
    for (int off = 16; off > 0; off >>= 1)
        lsum += __shfl_down(lsum, off, 32);
    if (lane == 0) atomicAdd(&acc[0], lsum);
}

// ---------------- finalize ----------------
__global__ void finalize_kernel(const float* __restrict__ acc, float* __restrict__ out) {
    if (threadIdx.x == 0) {
        float S = acc[0];
        // push = mean_b[(sum_b - G*W_HIGH)/((G-1)G+eps)*0.5]; affine per batch
        float push = (S - (float)BB * (float)GG * 1.0f)
                   / ((float)(GG - 1) * (float)GG + 1e-6f) * 0.5f / (float)BB;
        out[0] = push;
        out[1] = acc[1];   // pull already fully scaled
    }
}

extern "C" void kernel_launch(void* const* d_in, const int* in_sizes, int n_in,
                              void* d_out, int out_size, void* d_ws, size_t ws_size,
                              hipStream_t stream) {
    const float* pred   = (const float*)d_in[0];
    // d_in[1] = gt_inds (int32) — unused by the reference
    const float* anchor = (const float*)d_in[2];
    const float* offp   = (const float*)d_in[3];
    const float* gtb    = (const float*)d_in[4];
    float* out = (float*)d_out;

    float* ws   = (float*)d_ws;
    float* tags = ws;                                // B*G*TAGS = 16,384 floats
    float* acc  = tags + (size_t)BB * GG * TAGS;     // 2 scalars

    zero_acc_kernel<<<1, 32, 0, stream>>>(acc);
    fused_sample_pull_kernel<<<BB * GG, KK, 0, stream>>>(pred, anchor, offp, tags, acc);
    push_wmma_kernel<<<BB * 64, 32, 0, stream>>>(tags, gtb, acc);
    finalize_kernel<<<1, 1, 0, stream>>>(acc, out);
}